// CrossEntropyLoss_W_19396072309221
// MI455X (gfx1250) — compile-verified
//
#include <hip/hip_runtime.h>
#include <cstdint>

#define WAVES  8
#define WSZ    32
#define TPB    (WAVES * WSZ)        // 256 threads = 8 wave32
#define VEC    4                    // floats per lane per chunk (b128)
#define WCHUNK (WSZ * VEC)          // 128 floats per wave per chunk
#define CHUNK  (TPB * VEC)          // 1024 floats per workgroup chunk
#define DEPTH  8                    // async pipeline slots per wave
#define L2E    1.4426950408889634f  // log2(e)
#define RL2E   0.6931471805599453f  // ln(2)

__global__ __launch_bounds__(TPB)
void ce_loss_w_kernel(const float* __restrict__ logits,
                      const int*   __restrict__ target,
                      const float* __restrict__ weights,
                      float*       __restrict__ out,
                      int C)
{
    // 8 waves * 8 slots * 512B = 32KB streaming buffer (of 320KB/WGP)
    __shared__ __align__(16) float buf[WAVES * DEPTH * WCHUNK];
    __shared__ float redM[WAVES];
    __shared__ float redS[WAVES];

    const int row  = blockIdx.x;
    const int tid  = threadIdx.x;
    const int w    = tid >> 5;
    const int lane = tid & 31;

    const float* rowp = logits + (size_t)row * (size_t)C;

    // per-lane LDS byte address of this wave's slot 0 (low 32 bits = LDS offset)
    const unsigned lds0 =
        (unsigned)(uintptr_t)&buf[(w * DEPTH) * WCHUNK + lane * VEC];

    const int nC  = C / CHUNK;                 // full 1024-float chunks
    const int pre = (nC < DEPTH) ? nC : DEPTH;

    // ---- prologue: fill async pipeline (each wave loads its own sub-chunks)
    // Non-temporal: data is streamed exactly once; keep the 192MB L2 clean.
    for (int i = 0; i < pre; ++i) {
        unsigned long long g = (unsigned long long)(uintptr_t)
            (rowp + (size_t)i * CHUNK + w * WCHUNK + lane * VEC);
        unsigned la = lds0 + (unsigned)i * (WCHUNK * 4);
        asm volatile("global_load_async_to_lds_b128 %0, %1, off th:TH_LOAD_NT"
                     :: "v"(la), "v"(g) : "memory");
    }

    float m = -3.402823466e38f;   // running max (-FLT_MAX: stays finite)
    float s = 0.0f;               // running sum of 2^((x-m)*log2e)

    auto process = [&](int slot) {
        const float4 v = *reinterpret_cast<const float4*>(
            &buf[(w * DEPTH + slot) * WCHUNK + lane * VEC]);
        float cmax = fmaxf(fmaxf(v.x, v.y), fmaxf(v.z, v.w));
        if (cmax > m) {           // rare: rescale running sum
            s *= __builtin_amdgcn_exp2f((m - cmax) * L2E);
            m  = cmax;
        }
        s += __builtin_amdgcn_exp2f((v.x - m) * L2E);
        s += __builtin_amdgcn_exp2f((v.y - m) * L2E);
        s += __builtin_amdgcn_exp2f((v.z - m) * L2E);
        s += __builtin_amdgcn_exp2f((v.w - m) * L2E);
    };

    // ---- steady state: wait-oldest, consume, refill same slot
    const int main_end = nC - pre;            // >=0; >0 only when pre==DEPTH
    for (int c = 0; c < main_end; ++c) {
        asm volatile("s_wait_asynccnt 7" ::: "memory");   // DEPTH-1: chunk c landed
        const int slot = c & (DEPTH - 1);
        process(slot);
        asm volatile("s_wait_dscnt 0" ::: "memory");      // LDS reads done before reuse
        unsigned long long g = (unsigned long long)(uintptr_t)
            (rowp + (size_t)(c + DEPTH) * CHUNK + w * WCHUNK + lane * VEC);
        unsigned la = lds0 + (unsigned)slot * (WCHUNK * 4);
        asm volatile("global_load_async_to_lds_b128 %0, %1, off th:TH_LOAD_NT"
                     :: "v"(la), "v"(g) : "memory");
    }
    // ---- drain
    asm volatile("s_wait_asynccnt 0" ::: "memory");
    for (int c = main_end; c < nC; ++c) process(c & (DEPTH - 1));

    // ---- scalar tail (C % 1024 elements) via direct global loads
    const int tbase = nC * CHUNK;
    for (int i = tbase + tid; i < C; i += TPB) {
        float x = rowp[i];
        if (x > m) { s *= __builtin_amdgcn_exp2f((m - x) * L2E); m = x; }
        s += __builtin_amdgcn_exp2f((x - m) * L2E);
    }

    // ---- wave32 (m, s) reduction via lane shuffles
    #pragma unroll
    for (int off = 16; off > 0; off >>= 1) {
        float m2 = __shfl_xor(m, off, 32);
        float s2 = __shfl_xor(s, off, 32);
        float nm = fmaxf(m, m2);
        s = s  * __builtin_amdgcn_exp2f((m  - nm) * L2E)
          + s2 * __builtin_amdgcn_exp2f((m2 - nm) * L2E);
        m = nm;
    }
    if (lane == 0) { redM[w] = m; redS[w] = s; }
    __syncthreads();

    // ---- cross-wave combine + final answer
    if (tid == 0) {
        float fm = redM[0], fs = redS[0];
        #pragma unroll
        for (int i = 1; i < WAVES; ++i) {
            float m2 = redM[i], s2 = redS[i];
            float nm = fmaxf(fm, m2);
            fs = fs * __builtin_amdgcn_exp2f((fm - nm) * L2E)
               + s2 * __builtin_amdgcn_exp2f((m2 - nm) * L2E);
            fm = nm;
        }
        // logsumexp = fm + ln(fs);  fs = sum exp(x - fm)
        float lse = fm + __builtin_amdgcn_logf(fs) * RL2E;
        int   t   = target[row];
        float xt  = rowp[t];
        out[row]  = -weights[t] * (xt - lse);
    }
}

extern "C" void kernel_launch(void* const* d_in, const int* in_sizes, int n_in,
                              void* d_out, int out_size, void* d_ws, size_t ws_size,
                              hipStream_t stream) {
    const float* logits  = (const float*)d_in[0];
    const int*   target  = (const int*)  d_in[1];
    const float* weights = (const float*)d_in[2];
    float*       out     = (float*)d_out;

    const int N = in_sizes[1];                 // 8192 rows
    const int C = in_sizes[0] / N;             // 50257 classes

    ce_loss_w_kernel<<<N, TPB, 0, stream>>>(logits, target, weights, out, C);
}